// Graph_Decoder_Norm_Unpooling_1_32212254720655
// MI455X (gfx1250) — compile-verified
//
#include <hip/hip_runtime.h>
#include <hip/hip_bf16.h>

// ---------------------------------------------------------------------------
// Graph decoder: unpool -> 4x (GCNConv + LayerNorm + LeakyReLU) with 0.5-scaled
// skip accumulation.  f32 data -> use V_WMMA_F32_16X16X4_F32 for the X@W GEMMs.
// ---------------------------------------------------------------------------

#define N_NODES   50000
#define N_POOLED  25000
#define N_EDGES   800000
#define D_FEAT    64
#define N_LAYERS  4
#define LN_EPS    1e-5f
#define LRELU     0.01f

typedef __attribute__((ext_vector_type(2))) float v2f;
typedef __attribute__((ext_vector_type(8))) float v8f;

// ---------------------------------------------------------------------------
// init: cur = 0, h = 0, deg = 1 (self loops)
// ---------------------------------------------------------------------------
__global__ void k_init(float* __restrict__ cur, float* __restrict__ h,
                       float* __restrict__ deg) {
  int i = blockIdx.x * blockDim.x + threadIdx.x;
  if (i < N_NODES * D_FEAT) { cur[i] = 0.0f; h[i] = 0.0f; }
  if (i < N_NODES)          { deg[i] = 1.0f; }
}

// ---------------------------------------------------------------------------
// unpool: cur[idx[i], :] = x[i, :]
// ---------------------------------------------------------------------------
__global__ void k_unpool(const float* __restrict__ x, const int* __restrict__ idx,
                         float* __restrict__ cur) {
  int t = blockIdx.x * blockDim.x + threadIdx.x;
  if (t >= N_POOLED * D_FEAT) return;
  int i = t >> 6;            // pooled row
  int d = t & 63;            // feature
  cur[idx[i] * D_FEAT + d] = x[t];
}

// ---------------------------------------------------------------------------
// degree: deg[dst[e]] += 1
// ---------------------------------------------------------------------------
__global__ void k_degree(const int* __restrict__ dst, float* __restrict__ deg) {
  int e = blockIdx.x * blockDim.x + threadIdx.x;
  if (e < N_EDGES) atomicAdd(&deg[dst[e]], 1.0f);
}

// ---------------------------------------------------------------------------
// dinv: in-place deg -> rsqrt(max(deg,1))
// ---------------------------------------------------------------------------
__global__ void k_rsqrt(float* __restrict__ deg) {
  int i = blockIdx.x * blockDim.x + threadIdx.x;
  if (i < N_NODES) deg[i] = rsqrtf(fmaxf(deg[i], 1.0f));
}

// ---------------------------------------------------------------------------
// GEMM: Y[N_NODES,64] = X[N_NODES,64] @ W[64,64] using V_WMMA_F32_16X16X4_F32.
// One wave per 16-row tile; 4 v8f accumulators cover all 64 output cols.
// A 16x4 frag: lane L holds row (L&15), K = v + 2*(L>>4)   (v = 0,1)
// B 4x16 frag: lane L holds col (L&15), K = v + 2*(L>>4)
// C/D 16x16:   c[r] <-> row r + 8*(L>>4), col (L&15)
// ---------------------------------------------------------------------------
__global__ void k_gemm_wmma(const float* __restrict__ X, const float* __restrict__ W,
                            float* __restrict__ Y) {
  __shared__ float Wlds[D_FEAT * D_FEAT];          // 16 KB
  for (int i = threadIdx.x; i < D_FEAT * D_FEAT; i += blockDim.x)
    Wlds[i] = W[i];
  __syncthreads();

  const int lane    = threadIdx.x & 31;
  const int wave    = threadIdx.x >> 5;
  const int rowTile = blockIdx.x * (blockDim.x >> 5) + wave;
  if (rowTile >= N_NODES / 16) return;             // wave-uniform: EXEC all-1s inside

  const int row0 = rowTile * 16;
  const int m    = lane & 15;
  const int half = lane >> 4;

  v8f acc0 = {}, acc1 = {}, acc2 = {}, acc3 = {};
  const float* __restrict__ xrow = X + (size_t)(row0 + m) * D_FEAT + 2 * half;

  #pragma unroll
  for (int kk = 0; kk < D_FEAT; kk += 4) {
    v2f a;
    a.x = xrow[kk + 0];
    a.y = xrow[kk + 1];
    const int kb0 = (kk + 2 * half + 0) * D_FEAT + m;
    const int kb1 = (kk + 2 * half + 1) * D_FEAT + m;
    v2f b0, b1, b2, b3;
    b0.x = Wlds[kb0 +  0]; b0.y = Wlds[kb1 +  0];
    b1.x = Wlds[kb0 + 16]; b1.y = Wlds[kb1 + 16];
    b2.x = Wlds[kb0 + 32]; b2.y = Wlds[kb1 + 32];
    b3.x = Wlds[kb0 + 48]; b3.y = Wlds[kb1 + 48];
    acc0 = __builtin_amdgcn_wmma_f32_16x16x4_f32(false, a, false, b0, (short)0, acc0, false, false);
    acc1 = __builtin_amdgcn_wmma_f32_16x16x4_f32(false, a, false, b1, (short)0, acc1, false, false);
    acc2 = __builtin_amdgcn_wmma_f32_16x16x4_f32(false, a, false, b2, (short)0, acc2, false, false);
    acc3 = __builtin_amdgcn_wmma_f32_16x16x4_f32(false, a, false, b3, (short)0, acc3, false, false);
  }

  #pragma unroll
  for (int r = 0; r < 8; ++r) {
    float* yr = Y + (size_t)(row0 + r + 8 * half) * D_FEAT + m;
    yr[ 0] = acc0[r];
    yr[16] = acc1[r];
    yr[32] = acc2[r];
    yr[48] = acc3[r];
  }
}

// ---------------------------------------------------------------------------
// agg init: self-loop contribution + bias:  agg[i,d] = xw[i,d]*dinv[i]^2 + b[d]
// ---------------------------------------------------------------------------
__global__ void k_agg_init(const float* __restrict__ xw, const float* __restrict__ dinv,
                           const float* __restrict__ bias, float* __restrict__ agg) {
  int t = blockIdx.x * blockDim.x + threadIdx.x;
  if (t >= N_NODES * D_FEAT) return;
  int i = t >> 6;
  int d = t & 63;
  float dv = dinv[i];
  agg[t] = xw[t] * dv * dv + bias[d];
}

// ---------------------------------------------------------------------------
// edge scatter: one wave per edge, lane covers dims d and d+32.
// agg[dst,:] += xw[src,:] * dinv[src]*dinv[dst]
// ---------------------------------------------------------------------------
__global__ void k_edge_scatter(const int* __restrict__ src, const int* __restrict__ dst,
                               const float* __restrict__ dinv, const float* __restrict__ xw,
                               float* __restrict__ agg) {
  int e = blockIdx.x * (blockDim.x >> 5) + (threadIdx.x >> 5);
  if (e >= N_EDGES) return;
  int lane = threadIdx.x & 31;
  int s = src[e];
  int d = dst[e];
  float nrm = dinv[s] * dinv[d];
  const float* __restrict__ xs = xw + (size_t)s * D_FEAT;
  float* __restrict__ ad = agg + (size_t)d * D_FEAT;
  atomicAdd(&ad[lane],      xs[lane]      * nrm);
  atomicAdd(&ad[lane + 32], xs[lane + 32] * nrm);
}

// ---------------------------------------------------------------------------
// fused LayerNorm + LeakyReLU + skip accumulation.
// One wave per row; each wave32 lane owns channels (lane, lane+32).
// cur = act ; h += 0.5*act
// ---------------------------------------------------------------------------
__global__ void k_ln_act(const float* __restrict__ agg, const float* __restrict__ gamma,
                         const float* __restrict__ beta, float* __restrict__ cur,
                         float* __restrict__ h) {
  int row = blockIdx.x * (blockDim.x >> 5) + (threadIdx.x >> 5);
  if (row >= N_NODES) return;
  int lane = threadIdx.x & 31;
  size_t base = (size_t)row * D_FEAT;

  float v0 = agg[base + lane];
  float v1 = agg[base + lane + 32];

  float s = v0 + v1;
  #pragma unroll
  for (int off = 16; off > 0; off >>= 1) s += __shfl_xor(s, off, 32);
  float mu = s * (1.0f / 64.0f);

  float c0 = v0 - mu, c1 = v1 - mu;
  float q = c0 * c0 + c1 * c1;
  #pragma unroll
  for (int off = 16; off > 0; off >>= 1) q += __shfl_xor(q, off, 32);
  float rinv = rsqrtf(q * (1.0f / 64.0f) + LN_EPS);

  float y0 = c0 * rinv * gamma[lane]      + beta[lane];
  float y1 = c1 * rinv * gamma[lane + 32] + beta[lane + 32];
  float a0 = (y0 > 0.0f) ? y0 : LRELU * y0;
  float a1 = (y1 > 0.0f) ? y1 : LRELU * y1;

  cur[base + lane]      = a0;
  cur[base + lane + 32] = a1;
  h[base + lane]        += 0.5f * a0;
  h[base + lane + 32]   += 0.5f * a1;
}

// ---------------------------------------------------------------------------
// launch
// ---------------------------------------------------------------------------
extern "C" void kernel_launch(void* const* d_in, const int* in_sizes, int n_in,
                              void* d_out, int out_size, void* d_ws, size_t ws_size,
                              hipStream_t stream) {
  const float* x      = (const float*)d_in[0];               // [25000,64]
  const int*   eidx   = (const int*)  d_in[1];               // [2,800000]
  const int*   idx    = (const int*)  d_in[2];               // [25000]
  const float* Ws     = (const float*)d_in[3];               // [4,64,64]
  const float* bs     = (const float*)d_in[4];               // [4,64]
  const float* gammas = (const float*)d_in[5];               // [4,64]
  const float* betas  = (const float*)d_in[6];               // [4,64]
  float* h = (float*)d_out;                                  // [50000,64]

  const int* src = eidx;                                     // edge_index[0]
  const int* dst = eidx + N_EDGES;                           // edge_index[1]

  // workspace carve-up (~38.6 MB)
  float* cur  = (float*)d_ws;                                // 50000*64
  float* xw   = cur + (size_t)N_NODES * D_FEAT;              // 50000*64
  float* agg  = xw  + (size_t)N_NODES * D_FEAT;              // 50000*64
  float* dinv = agg + (size_t)N_NODES * D_FEAT;              // 50000 (deg -> dinv)

  const int TPB = 256;
  const int nFeat   = N_NODES * D_FEAT;                      // 3.2M
  const int gFeat   = (nFeat + TPB - 1) / TPB;               // 12500
  const int gUnpool = (N_POOLED * D_FEAT + TPB - 1) / TPB;   // 6250
  const int gDeg    = (N_EDGES + TPB - 1) / TPB;             // 3125
  const int gNodes  = (N_NODES + TPB - 1) / TPB;             // 196
  const int wavesPB = TPB / 32;                              // 8 waves/block
  const int gGemm   = (N_NODES / 16 + wavesPB - 1) / wavesPB;    // 391
  const int gEdge   = (N_EDGES + wavesPB - 1) / wavesPB;         // 100000
  const int gRows   = (N_NODES + wavesPB - 1) / wavesPB;         // 6250

  k_init  <<<gFeat,   TPB, 0, stream>>>(cur, h, dinv);
  k_unpool<<<gUnpool, TPB, 0, stream>>>(x, idx, cur);
  k_degree<<<gDeg,    TPB, 0, stream>>>(dst, dinv);
  k_rsqrt <<<gNodes,  TPB, 0, stream>>>(dinv);

  for (int l = 0; l < N_LAYERS; ++l) {
    const float* W = Ws     + (size_t)l * D_FEAT * D_FEAT;
    const float* b = bs     + (size_t)l * D_FEAT;
    const float* g = gammas + (size_t)l * D_FEAT;
    const float* be= betas  + (size_t)l * D_FEAT;

    k_gemm_wmma   <<<gGemm, TPB, 0, stream>>>(cur, W, xw);
    k_agg_init    <<<gFeat, TPB, 0, stream>>>(xw, dinv, b, agg);
    k_edge_scatter<<<gEdge, TPB, 0, stream>>>(src, dst, dinv, xw, agg);
    k_ln_act      <<<gRows, TPB, 0, stream>>>(agg, g, be, cur, h);
  }
}